// CEMA_59485297049751
// MI455X (gfx1250) — compile-verified
//
#include <hip/hip_runtime.h>
#include <math.h>

#define S_LEN 8192
#define D_LEN 1024
#define B_LEN 4
#define NDIM  16
#define CHUNK 128
#define NCHUNK (S_LEN / CHUNK)   /* 64 */
#define TWO_PI 6.283185307179586f

typedef __attribute__((ext_vector_type(2))) float v2f;
typedef __attribute__((ext_vector_type(8))) float v8f;

__device__ __forceinline__ float softplus_f(float z) {
  // stable softplus: max(z,0) + log(1 + exp(-|z|)), fast hw exp/log
  float e = __expf(-fabsf(z));
  return fmaxf(z, 0.0f) + __logf(1.0f + e);
}

__device__ __forceinline__ float tanh_f(float z) {
#if __has_builtin(__builtin_amdgcn_tanhf)
  return __builtin_amdgcn_tanhf(z);        // v_tanh_f32 (gfx1250 trans unit)
#elif __has_builtin(__builtin_amdgcn_tanh_f32)
  return __builtin_amdgcn_tanh_f32(z);
#else
  return tanhf(z);
#endif
}

// ---------------------------------------------------------------------------
// K_pe: pe[s,d] = softplus( sum_n tanh(alpha[n,d]*sin(s*t_n)+beta[n,d]*cos(s*t_n)) )
// One block per s-row; sin/cos (uniform over d) hoisted into LDS.
// ---------------------------------------------------------------------------
__global__ void pe_kernel(const float* __restrict__ alpha,
                          const float* __restrict__ beta,
                          float* __restrict__ pe) {
  const int s = blockIdx.x;
  const int t = threadIdx.x;             // 128 threads
  __shared__ float sv[NDIM], cv[NDIM];
  if (t < NDIM) {
    float ang = (float)s * ((TWO_PI / 15.0f) * (float)t);
    sv[t] = sinf(ang);
    cv[t] = cosf(ang);
  }
  __syncthreads();
  for (int d = t; d < D_LEN; d += 128) {
    float acc = 0.0f;
#pragma unroll
    for (int n = 0; n < NDIM; ++n) {
      acc += tanh_f(fmaf(alpha[n * D_LEN + d], sv[n], beta[n * D_LEN + d] * cv[n]));
    }
    pe[(size_t)s * D_LEN + d] = softplus_f(acc);
  }
}

// ---------------------------------------------------------------------------
// K1: per-(b,chunk,d) sum of softplus(x * softplus(omega)) over the chunk.
// ---------------------------------------------------------------------------
__global__ void partial_kernel(const float* __restrict__ x,
                               const float* __restrict__ omega,
                               float* __restrict__ psum) {
  const int d = blockIdx.x * 256 + threadIdx.x;
  const int c = blockIdx.y;
  const int b = blockIdx.z;
  const float wact = softplus_f(omega[d]);
  const float* xp = x + ((size_t)b * S_LEN + (size_t)c * CHUNK) * D_LEN + d;
  float sum = 0.0f;
#pragma unroll 4
  for (int s = 0; s < CHUNK; ++s) {
    sum += softplus_f(xp[(size_t)s * D_LEN] * wact);
  }
  psum[((size_t)(b * NCHUNK + c)) * D_LEN + d] = sum;
}

// ---------------------------------------------------------------------------
// K2: in-place exclusive scan of the 64 chunk totals per (b,d) column.
// ---------------------------------------------------------------------------
__global__ void scan_kernel(float* __restrict__ psum) {
  const int gid = blockIdx.x * 256 + threadIdx.x;   // 4096 columns
  const int b = gid >> 10;
  const int d = gid & (D_LEN - 1);
  float run = 0.0f;
  for (int c = 0; c < NCHUNK; ++c) {
    size_t i = ((size_t)(b * NCHUNK + c)) * D_LEN + d;
    float v = psum[i];
    psum[i] = run;
    run += v;
  }
}

// ---------------------------------------------------------------------------
// K3: fused scan + output. One wave per (chunk, 16-column tile, batch).
// Intra-16 prefix = L(16x16 lower-tri ones) @ X via 4 chained
// V_WMMA_F32_16X16X4_F32; carry rides in the C accumulator.
// All global accesses use one base pointer per tile + constant IOFFSETs,
// and pe loads are issued before the WMMA chain to hide latency.
// ---------------------------------------------------------------------------
__global__ void __launch_bounds__(32)
cema_kernel(const float* __restrict__ x,
            const float* __restrict__ omega,
            const float* __restrict__ gamma,
            const float* __restrict__ pe,
            const float* __restrict__ psum,
            float* __restrict__ out) {
  const int c  = blockIdx.x;           // chunk
  const int d0 = blockIdx.y * 16;      // column tile
  const int b  = blockIdx.z;           // batch
  const int lane = threadIdx.x;        // wave32
  const int n = lane & 15;             // column within tile / matrix N (and A row M)
  const int h = lane >> 4;             // half-wave select

  const int colD = d0 + n;
  const float wact = softplus_f(omega[colD]);
  const float gact = softplus_f(gamma[colD]);
  float carry = psum[((size_t)(b * NCHUNK + c)) * D_LEN + colD];

  // A operands: L[:,4m+2h] and L[:,4m+2h+1] at row n (ISA 16x4 f32 A layout)
  v2f amat[4];
#pragma unroll
  for (int m = 0; m < 4; ++m) {
    int klo = 4 * m + 2 * h;
    amat[m][0] = (klo     <= n) ? 1.0f : 0.0f;
    amat[m][1] = (klo + 1 <= n) ? 1.0f : 0.0f;
  }

  // column-major xs tile, stride 18 words (bank-conflict-free, b64-aligned)
  __shared__ __align__(16) float xs[16 * 18];

  // per-tile base pointers; lane-fixed part (row 8h, column n) folded in so
  // the v-loop offsets are compile-time constants -> global IOFFSET encoding
  const int sChunk = c * CHUNK;
  const float* xP  = x   + ((size_t)b * S_LEN + sChunk + 8 * h) * D_LEN + colD;
  const float* peP = pe  + ((size_t)sChunk + 8 * h) * D_LEN + colD;
  float*       oP  = out + ((size_t)b * S_LEN + sChunk + 8 * h) * D_LEN + colD;

  for (int t8 = 0; t8 < CHUNK / 16; ++t8) {
    // ---- phase 1: batched x loads (rows v+8h, column n) -------------------
    float xv[8];
#pragma unroll
    for (int v = 0; v < 8; ++v) xv[v] = xP[(size_t)v * D_LEN];

    // ---- phase 2: batched pe loads (latency hidden by softplus + WMMA) ----
    float pv[8];
#pragma unroll
    for (int v = 0; v < 8; ++v) pv[v] = peP[(size_t)v * D_LEN];

    if (t8 < CHUNK / 16 - 1) {
      __builtin_prefetch(&xP[(size_t)16 * D_LEN], 0, 3);
    }

    // ---- phase 3: softplus, stage into LDS column-major -------------------
#pragma unroll
    for (int v = 0; v < 8; ++v) {
      xs[n * 18 + (v + 8 * h)] = softplus_f(xv[v] * wact);
    }
    __syncthreads();   // single-wave WG: S_NOP, but orders DS ops

    // ---- phase 4: C init = running carry; L @ X via 4 chained WMMAs -------
    v8f acc;
#pragma unroll
    for (int i = 0; i < 8; ++i) acc[i] = carry;

#pragma unroll
    for (int m = 0; m < 4; ++m) {
      int klo = 4 * m + 2 * h;
      v2f bv = *reinterpret_cast<const v2f*>(&xs[n * 18 + klo]);
      acc = __builtin_amdgcn_wmma_f32_16x16x4_f32(
          false, amat[m], false, bv, (short)0, acc, false, false);
    }

    // new carry = row 15 of this column (vgpr7 of upper half-wave)
    carry = __shfl(acc[7], 16 + n, 32);

    // ---- phase 5: combine + batched stores --------------------------------
    float ov[8];
#pragma unroll
    for (int v = 0; v < 8; ++v) ov[v] = fmaf(gact * pv[v], acc[v], xv[v]);
#pragma unroll
    for (int v = 0; v < 8; ++v) oP[(size_t)v * D_LEN] = ov[v];

    __syncthreads();

    xP  += (size_t)16 * D_LEN;
    peP += (size_t)16 * D_LEN;
    oP  += (size_t)16 * D_LEN;
  }
}

// ---------------------------------------------------------------------------
extern "C" void kernel_launch(void* const* d_in, const int* in_sizes, int n_in,
                              void* d_out, int out_size, void* d_ws, size_t ws_size,
                              hipStream_t stream) {
  (void)in_sizes; (void)n_in; (void)out_size; (void)ws_size;
  const float* x     = (const float*)d_in[0];
  const float* omega = (const float*)d_in[1];
  const float* alpha = (const float*)d_in[2];
  const float* beta  = (const float*)d_in[3];
  const float* gamma = (const float*)d_in[4];
  float* out  = (float*)d_out;
  float* pe   = (float*)d_ws;                       // S*D floats = 32 MB
  float* psum = pe + (size_t)S_LEN * D_LEN;         // B*NCHUNK*D floats = 1 MB

  pe_kernel<<<dim3(S_LEN), 128, 0, stream>>>(alpha, beta, pe);
  partial_kernel<<<dim3(D_LEN / 256, NCHUNK, B_LEN), 256, 0, stream>>>(x, omega, psum);
  scan_kernel<<<dim3((B_LEN * D_LEN) / 256), 256, 0, stream>>>(psum);
  cema_kernel<<<dim3(NCHUNK, D_LEN / 16, B_LEN), 32, 0, stream>>>(
      x, omega, gamma, pe, psum, out);
}